// GRNNmodel_40192303956632
// MI455X (gfx1250) — compile-verified
//
#include <hip/hip_runtime.h>

// ---------------------------------------------------------------------------
// Kalman/Riccati scan, B=256 batches, T=16384 steps, 2-D state.
//  * cov chain is batch-invariant -> one serial kernel emits M_t / X_t.
//  * linear time-varying state scan -> chunked decomposition:
//      psi (zero-state responses, staged in d_out), Phi (chunk prefixes),
//      x_start (short sequential combine), final GEMM-per-chunk with
//      V_WMMA_F32_16X16X4_F32 (psi preloaded as the C accumulator, in-place).
//  * psi kernel stages the chunk's 4KB M/X coefficient tile into LDS with the
//    Tensor Data Mover (tensor_load_to_lds + s_wait_tensorcnt), then all 8
//    waves read it via broadcast DS loads.
// ---------------------------------------------------------------------------

#define T_LEN 16384
#define B_N   256
#define L_CHK 128
#define G_CHK 128

static constexpr float GAMMA_C = 0.3f;
static constexpr float G2_C    = 0.15f;                 // gamma/2
static constexpr float DCON    = 1.55f;                 // gamma*(nbar+.5)+lambda
static constexpr float CCON    = 1.7888543819998317f;   // sqrt(4*eta*lambda)
static constexpr float DT_C    = 1e-3f;

typedef float        v2f __attribute__((ext_vector_type(2)));
typedef float        v8f __attribute__((ext_vector_type(8)));
typedef unsigned int v4u __attribute__((ext_vector_type(4)));
typedef int          v4i __attribute__((ext_vector_type(4)));
typedef int          v8i __attribute__((ext_vector_type(8)));

// ---- Kernel 1: serial covariance (Riccati) chain; emits interleaved M,X ---
__global__ void k_covchain(const float* __restrict__ cov0,
                           const float* __restrict__ coeffs,
                           float* __restrict__ wsMX) {
  if (threadIdx.x != 0 || blockIdx.x != 0) return;
  const float w = coeffs[0];
  float p = cov0[0], q = cov0[1], r = cov0[3];       // cov = [[p,q],[q,r]]
  for (int t = 0; t < T_LEN; ++t) {
    float xp = fmaf(CCON, p, DCON);                  // xicov = c*cov + d*I
    float xq = CCON * q;
    float xr = fmaf(CCON, r, DCON);
    float4 M;
    M.x = fmaf(DT_C, fmaf(-CCON, xp, -G2_C), 1.0f);  // 1+dt(-g/2 - c*xp)
    M.y = DT_C * ( w - CCON * xq);
    M.z = DT_C * (-w - CCON * xq);
    M.w = fmaf(DT_C, fmaf(-CCON, xr, -G2_C), 1.0f);
    float4 X; X.x = xp; X.y = xq; X.z = xq; X.w = xr;
    ((float4*)wsMX)[(size_t)t * 2]     = M;          // 32B record per step
    ((float4*)wsMX)[(size_t)t * 2 + 1] = X;
    // Riccati: cov' = cov + dt(Acov + covA^T + dI - xicov*xicov^T)
    float pd = fmaf(-GAMMA_C, p, fmaf(2.0f * w, q, DCON)) - fmaf(xp, xp, xq * xq);
    float qd = fmaf(-GAMMA_C, q, w * (r - p)) - xq * (xp + xr);
    float rd = fmaf(-GAMMA_C, r, fmaf(-2.0f * w, q, DCON)) - fmaf(xq, xq, xr * xr);
    p = fmaf(DT_C, pd, p);
    q = fmaf(DT_C, qd, q);
    r = fmaf(DT_C, rd, r);
  }
}

// ---- Kernel 2: chunk prefix products Phi_k and chunk transfer M^_g --------
__global__ void k_phi(const float* __restrict__ wsMX,
                      float* __restrict__ wsPhi, float* __restrict__ wsMhat) {
  int g = blockIdx.x * blockDim.x + threadIdx.x;
  if (g >= G_CHK) return;
  float p00 = 1.f, p01 = 0.f, p10 = 0.f, p11 = 1.f;   // Phi_0 = I
  int t0 = g * L_CHK;
  for (int k = 0; k < L_CHK; ++k) {
    float4 P; P.x = p00; P.y = p01; P.z = p10; P.w = p11;
    ((float4*)wsPhi)[g * L_CHK + k] = P;
    float4 M = ((const float4*)wsMX)[(size_t)(t0 + k) * 2];
    float n00 = fmaf(M.x, p00, M.y * p10);
    float n01 = fmaf(M.x, p01, M.y * p11);
    float n10 = fmaf(M.z, p00, M.w * p10);
    float n11 = fmaf(M.z, p01, M.w * p11);
    p00 = n00; p01 = n01; p10 = n10; p11 = n11;
  }
  float4 P; P.x = p00; P.y = p01; P.z = p10; P.w = p11;
  ((float4*)wsMhat)[g] = P;
}

// ---- Kernel 3: per-(batch,chunk) zero-state responses psi -> d_out --------
// TDM stages the chunk's M/X tile (L_CHK * 32B = 4KB) into LDS once per block.
__global__ void k_psi(const float* __restrict__ dy,
                      const float* __restrict__ wsMX,
                      float* __restrict__ outPsi,   // = d_out (staging)
                      float* __restrict__ wsB) {
  __shared__ float ldsMX[L_CHK * 8];   // 4KB coefficient tile

  int b  = threadIdx.x;                // 0..255
  int g  = blockIdx.x;                 // 0..127
  int t0 = g * L_CHK;

  // --- Tensor Data Mover: global wsMX[t0..t0+L) -> LDS (wave 0 issues) ----
  if (threadIdx.x < 32) {
    unsigned long long ga = (unsigned long long)(wsMX) + (unsigned long long)t0 * 32ull;
    unsigned ldsa = (unsigned)(unsigned long long)(uintptr_t)&ldsMX[0];
    // D# group0: count=1 | lds_addr[63:32] | global_addr[120:64] | type=2
    v4u g0;
    g0[0] = 1u;                                   // count=1, user mode
    g0[1] = ldsa;                                 // LDS byte address
    g0[2] = (unsigned)(ga & 0xFFFFFFFFull);       // global_addr[31:0]
    g0[3] = (unsigned)((ga >> 32) & 0x01FFFFFFull) | 0x80000000u; // ga[56:32], type=2
    // D# group1: 1-D tile: data_size=3 (8B), tensor_dim0=tile_dim0=512,
    //            tensor_dim0_stride=512 (elements), everything else 0.
    v8i g1;
    g1[0] = (3 << 16);                            // wg_mask=0, data_size=8B
    g1[1] = (int)(512u << 16);                    // tensor_dim0[15:0] << 16
    g1[2] = 0;                                    // tensor_dim0 hi / tensor_dim1 lo
    g1[3] = (int)(512u << 16);                    // tile_dim0 = 512
    g1[4] = 0;                                    // tile_dim1/2 unused
    g1[5] = 512;                                  // tensor_dim0_stride lo32
    g1[6] = 0;
    g1[7] = 0;
    v4i gz4; gz4[0] = 0; gz4[1] = 0; gz4[2] = 0; gz4[3] = 0;  // groups 2/3 (<=2D)
    v8i gz8; gz8[0] = 0; gz8[1] = 0; gz8[2] = 0; gz8[3] = 0;
             gz8[4] = 0; gz8[5] = 0; gz8[6] = 0; gz8[7] = 0;
    __builtin_amdgcn_tensor_load_to_lds(g0, g1, gz4, gz4, gz8, 0);
    __builtin_amdgcn_s_wait_tensorcnt(0);
  }
  __syncthreads();

  const float* dyp = dy     + ((size_t)b * T_LEN + t0) * 2;
  float*       op  = outPsi + ((size_t)b * T_LEN + t0) * 2;
  const float4* mx = (const float4*)ldsMX;

  float x0 = 0.f, x1 = 0.f;
  for (int ko = 0; ko < L_CHK; ko += 16) {
    __builtin_prefetch(dyp + (ko + 16) * 2, 0, 1);   // global_prefetch_b8
    for (int ki = 0; ki < 16; ++ki) {
      int k = ko + ki;
      float4 M  = mx[k * 2];                         // ds_load_b128 (broadcast)
      float4 X  = mx[k * 2 + 1];
      float2 d2 = ((const float2*)dyp)[k];
      float2 ps; ps.x = x0; ps.y = x1;
      ((float2*)op)[k] = ps;                         // psi_k (pre-update)
      float n0 = fmaf(M.x, x0, M.y * x1) + fmaf(X.x, d2.x, X.y * d2.y);
      float n1 = fmaf(M.z, x0, M.w * x1) + fmaf(X.y, d2.x, X.w * d2.y);
      x0 = n0; x1 = n1;
    }
  }
  wsB[((size_t)b * G_CHK + g) * 2]     = x0;         // b_g = psi_L
  wsB[((size_t)b * G_CHK + g) * 2 + 1] = x1;
}

// ---- Kernel 4: chunk-start states via short sequential combine ------------
__global__ void k_xstart(const float* __restrict__ x0in,
                         const float* __restrict__ wsMhat,
                         const float* __restrict__ wsB,
                         float* __restrict__ wsXs) {
  int b = threadIdx.x;                 // one block of 256
  float x0 = x0in[b * 2], x1 = x0in[b * 2 + 1];
  for (int g = 0; g < G_CHK; ++g) {
    wsXs[((size_t)b * G_CHK + g) * 2]     = x0;
    wsXs[((size_t)b * G_CHK + g) * 2 + 1] = x1;
    float4 M = ((const float4*)wsMhat)[g];
    float b0 = wsB[((size_t)b * G_CHK + g) * 2];
    float b1 = wsB[((size_t)b * G_CHK + g) * 2 + 1];
    float n0 = fmaf(M.x, x0, fmaf(M.y, x1, b0));
    float n1 = fmaf(M.z, x0, fmaf(M.w, x1, b1));
    x0 = n0; x1 = n1;
  }
}

// ---- Kernel 5: out = c*dt*(Phi_k @ xstart + psi)  via WMMA f32 16x16x4 ----
// A (16x4): rows = 16 batches, K=0,1 carry xstart, K=2,3 zero padding.
// B (4x16): cols = 16 flattened (k,i) outputs, rows K=0,1 carry Phi_k[i][:].
// C/D (16x16): row = vgpr + 8*(lane>=16), col = lane&15. C preloaded with psi.
__global__ void k_out_wmma(const float* __restrict__ wsXs,
                           const float* __restrict__ wsPhi,
                           float* __restrict__ out) {
  int g    = blockIdx.x;                       // chunk
  int wave = threadIdx.x >> 5;                 // 0..3
  int lane = threadIdx.x & 31;
  int mtile = blockIdx.y * 4 + wave;           // 0..15 (batch tile)
  int l15  = lane & 15;
  float msk = (lane < 16) ? 1.0f : 0.0f;       // arithmetic mask: EXEC stays all-1
  int brow = mtile * 16 + l15;

  v2f A;
  A[0] = wsXs[((size_t)brow * G_CHK + g) * 2]     * msk;
  A[1] = wsXs[((size_t)brow * G_CHK + g) * 2 + 1] * msk;

  const float cdt = CCON * DT_C;
  int t0 = g * L_CHK;
  int crow_off = (lane >> 4) * 8;

  for (int n = 0; n < 16; ++n) {
    int ncol = n * 16 + l15;                   // column within chunk = 2k+i
    int k = ncol >> 1;
    int i = ncol & 1;
    const float* phirow = wsPhi + ((size_t)g * L_CHK + k) * 4 + i * 2;
    v2f Bm;
    Bm[0] = phirow[0] * msk;                   // Phi_k[i][0]
    Bm[1] = phirow[1] * msk;                   // Phi_k[i][1]

    size_t colflat = (size_t)t0 * 2 + (size_t)n * 16 + l15;
    v8f C;
#pragma unroll
    for (int v = 0; v < 8; ++v) {
      int br = mtile * 16 + v + crow_off;
      C[v] = out[(size_t)br * (T_LEN * 2) + colflat];   // psi tile
    }
    v8f D = __builtin_amdgcn_wmma_f32_16x16x4_f32(
        false, A, false, Bm, (short)0, C, false, false);
#pragma unroll
    for (int v = 0; v < 8; ++v) {
      int br = mtile * 16 + v + crow_off;
      out[(size_t)br * (T_LEN * 2) + colflat] = cdt * D[v];
    }
  }
}

// ---------------------------------------------------------------------------
extern "C" void kernel_launch(void* const* d_in, const int* in_sizes, int n_in,
                              void* d_out, int out_size, void* d_ws, size_t ws_size,
                              hipStream_t stream) {
  const float* dy     = (const float*)d_in[0];   // [256,16384,2]
  const float* x0     = (const float*)d_in[1];   // [256,2]
  const float* cov0   = (const float*)d_in[2];   // [256,2,2]
  const float* coeffs = (const float*)d_in[3];   // [1,1]
  float* out = (float*)d_out;                    // [256,16384,2]

  // workspace layout (floats), ~1.32 MB total
  float* ws     = (float*)d_ws;
  float* wsMX   = ws;                                  // T*8 (M,X interleaved)
  float* wsPhi  = wsMX   + (size_t)T_LEN * 8;          // G*L*4
  float* wsMhat = wsPhi  + (size_t)G_CHK * L_CHK * 4;  // G*4
  float* wsB    = wsMhat + (size_t)G_CHK * 4;          // B*G*2
  float* wsXs   = wsB    + (size_t)B_N * G_CHK * 2;    // B*G*2

  // 1) serial Riccati chain -> interleaved M_t, X_t records
  k_covchain<<<dim3(1), dim3(32), 0, stream>>>(cov0, coeffs, wsMX);
  // 2) chunk prefix products (batch-independent)
  k_phi<<<dim3(1), dim3(G_CHK), 0, stream>>>(wsMX, wsPhi, wsMhat);
  // 3) zero-state responses psi -> staged in d_out; chunk sums -> wsB
  k_psi<<<dim3(G_CHK), dim3(B_N), 0, stream>>>(dy, wsMX, out, wsB);
  // 4) chunk-start states
  k_xstart<<<dim3(1), dim3(B_N), 0, stream>>>(x0, wsMhat, wsB, wsXs);
  // 5) in-place combine with WMMA: out = c*dt*(Phi@xstart + psi)
  k_out_wmma<<<dim3(G_CHK, 4), dim3(128), 0, stream>>>(wsXs, wsPhi, out);
}